// ZPriorLoss_84421877170330
// MI455X (gfx1250) — compile-verified
//
#include <hip/hip_runtime.h>
#include <math.h>

typedef __attribute__((ext_vector_type(2))) float v2f;
typedef __attribute__((ext_vector_type(8))) float v8f;

#define NROWS 8192
#define DDIM 64
#define TILE 16
#define COL_CHUNKS 8        // gridDim.x
#define TILES_PER_CHUNK 64  // 64*16 = 1024 cols per chunk
#define ROW_BLOCKS 64       // gridDim.y : 64 * 128 = 8192 rows
#define WAVES_PER_BLOCK 8   // 256 threads
#define PARTIALS_PER_MAT (COL_CHUNKS * ROW_BLOCKS)  // 512

// d_ws layout (floats):
//   [0, 524288)        xbuf  : normalized x, [8192][64]
//   [524288, 532480)   xnorm : ||x_i||^2
//   [532480, 540672)   ynorm : ||y_j||^2
//   [540672, 542208)   partials [3][512]
#define WS_XBUF   0
#define WS_XNORM  (NROWS * DDIM)
#define WS_YNORM  (WS_XNORM + NROWS)
#define WS_PART   (WS_YNORM + NROWS)

// ---------------- Stage 1: normalize x, compute row norms ----------------
__global__ __launch_bounds__(256) void mmd_prep(
    const float* __restrict__ zseq, const float* __restrict__ mean_,
    const float* __restrict__ std_, const float* __restrict__ ypr,
    float* __restrict__ xbuf, float* __restrict__ xnorm, float* __restrict__ ynorm)
{
  const int lane = threadIdx.x & 31;
  const int wave = threadIdx.x >> 5;
  const int row  = blockIdx.x * WAVES_PER_BLOCK + wave;
  const int k    = lane * 2;

  float p;
  if (blockIdx.y == 0) {
    const float* src = zseq + (size_t)row * DDIM + k;
    const float a = (src[0] - mean_[k])     / std_[k];
    const float b = (src[1] - mean_[k + 1]) / std_[k + 1];
    float* dst = xbuf + (size_t)row * DDIM + k;
    dst[0] = a;
    dst[1] = b;
    p = a * a + b * b;
  } else {
    const float* src = ypr + (size_t)row * DDIM + k;
    const float a = src[0], b = src[1];
    p = a * a + b * b;
  }
#pragma unroll
  for (int off = 16; off > 0; off >>= 1) p += __shfl_down(p, off, 32);
  if (lane == 0) {
    if (blockIdx.y == 0) xnorm[row] = p;
    else                 ynorm[row] = p;
  }
}

// ---------------- Stage 2: WMMA Gram tiles + fused RBF epilogue ----------------
// WHICH: 0 = kxx (exclude diag), 1 = kyy (exclude diag), 2 = kxy
template <int WHICH>
__global__ __launch_bounds__(256) void mmd_rbf_tiles(
    const float* __restrict__ abuf,  const float* __restrict__ bbuf,
    const float* __restrict__ anorm, const float* __restrict__ bnorm,
    float* __restrict__ partials)    // [512] for this matrix
{
  constexpr bool EXCL_DIAG = (WHICH != 2);

  const int lane = threadIdx.x & 31;
  const int wave = threadIdx.x >> 5;

  const int rowBase = blockIdx.y * (WAVES_PER_BLOCK * TILE) + wave * TILE;
  const int colChunkBase = blockIdx.x * (TILES_PER_CHUNK * TILE);

  const int lo = lane & 15;  // row (A) / col (B) within tile
  const int hi = lane >> 4;  // selects K pair within group of 4
  const int kb = hi * 2;

  // ---- A tile (16 rows x 64 K) in registers; v_wmma_f32_16x16x4_f32 layout ----
  const float* arow = abuf + (size_t)(rowBase + lo) * DDIM + kb;
  v2f af[16];
#pragma unroll
  for (int kk = 0; kk < 16; ++kk) {
    af[kk].x = arow[4 * kk];
    af[kk].y = arow[4 * kk + 1];
  }

  // Row norms for this lane's C slots: C VGPR i -> row (i + 8*hi)
  float xn8[8];
#pragma unroll
  for (int i = 0; i < 8; ++i) xn8[i] = anorm[rowBase + i + 8 * hi];

  float lsum = 0.0f;

  // B-tile loader (prefetch target selected by caller for double buffering)
  auto loadB = [&](v2f* bf, int tileIdx) {
    const int colBase = colChunkBase + tileIdx * TILE;
    const float* brow = bbuf + (size_t)(colBase + lo) * DDIM + kb;
#pragma unroll
    for (int kk = 0; kk < 16; ++kk) {
      bf[kk].x = brow[4 * kk];
      bf[kk].y = brow[4 * kk + 1];
    }
  };

  // 16-deep WMMA chain + fused RBF epilogue for one 16x16 tile
  auto doTile = [&](const v2f* bf, int tileIdx) {
    v8f c = {};
#pragma unroll
    for (int kk = 0; kk < 16; ++kk) {
      c = __builtin_amdgcn_wmma_f32_16x16x4_f32(
          false, af[kk], false, bf[kk], (short)0, c, false, false);
    }
    const int   colBase = colChunkBase + tileIdx * TILE;
    const float yn      = bnorm[colBase + lo];
    const int   colg    = colBase + lo;
#pragma unroll
    for (int i = 0; i < 8; ++i) {
      float d2 = fmaf(-2.0f, c[i], xn8[i] + yn);
      d2 = fmaxf(d2, 0.0f);
      float v = __expf(d2 * (-1.0f / 128.0f));
      if (EXCL_DIAG) {
        const int rowg = rowBase + i + 8 * hi;
        if (rowg == colg) v = 0.0f;  // implements (sum - trace)
      }
      lsum += v;
    }
  };

  // ---- column-tile loop, software-pipelined with ping-pong B buffers ----
  v2f bufA[16], bufB[16];
  loadB(bufA, 0);
  for (int t = 0; t < TILES_PER_CHUNK; t += 2) {
    loadB(bufB, (t + 1) & (TILES_PER_CHUNK - 1));  // prefetch next tile
    doTile(bufA, t);
    loadB(bufA, (t + 2) & (TILES_PER_CHUNK - 1));  // wraps at tail: safe, redundant
    doTile(bufB, t + 1);
  }

  // ---- wave reduce, block reduce, one deterministic partial per block ----
#pragma unroll
  for (int off = 16; off > 0; off >>= 1) lsum += __shfl_down(lsum, off, 32);

  __shared__ float ssum[WAVES_PER_BLOCK];
  if (lane == 0) ssum[wave] = lsum;
  __syncthreads();
  if (threadIdx.x == 0) {
    float s = 0.0f;
#pragma unroll
    for (int w = 0; w < WAVES_PER_BLOCK; ++w) s += ssum[w];
    partials[blockIdx.y * COL_CHUNKS + blockIdx.x] = s;
  }
}

// ---------------- Stage 3: deterministic final reduction ----------------
__global__ __launch_bounds__(256) void mmd_finalize(const float* __restrict__ partials,
                                                    float* __restrict__ out)
{
  __shared__ float s0[256], s1[256], s2[256];
  const int t = threadIdx.x;
  float a0 = 0.0f, a1 = 0.0f, a2 = 0.0f;
  for (int i = t; i < PARTIALS_PER_MAT; i += 256) {
    a0 += partials[0 * PARTIALS_PER_MAT + i];
    a1 += partials[1 * PARTIALS_PER_MAT + i];
    a2 += partials[2 * PARTIALS_PER_MAT + i];
  }
  s0[t] = a0; s1[t] = a1; s2[t] = a2;
  __syncthreads();
  for (int off = 128; off > 0; off >>= 1) {
    if (t < off) { s0[t] += s0[t + off]; s1[t] += s1[t + off]; s2[t] += s2[t + off]; }
    __syncthreads();
  }
  if (t == 0) {
    const float n = (float)NROWS;
    const float inv_pair = 1.0f / (n * (n - 1.0f));
    float mmd = s0[0] * inv_pair + s1[0] * inv_pair - 2.0f * s2[0] / (n * n);
    out[0] = fmaxf(mmd, 0.0f);
  }
}

extern "C" void kernel_launch(void* const* d_in, const int* in_sizes, int n_in,
                              void* d_out, int out_size, void* d_ws, size_t ws_size,
                              hipStream_t stream) {
  const float* z_seq   = (const float*)d_in[0];  // [16,512,64] -> [8192][64]
  const float* p_mean  = (const float*)d_in[1];  // [64]
  const float* p_std   = (const float*)d_in[2];  // [64]
  const float* z_prior = (const float*)d_in[3];  // [8192][64]

  float* ws       = (float*)d_ws;
  float* xbuf     = ws + WS_XBUF;
  float* xnorm    = ws + WS_XNORM;
  float* ynorm    = ws + WS_YNORM;
  float* partials = ws + WS_PART;

  mmd_prep<<<dim3(NROWS / WAVES_PER_BLOCK, 2), 256, 0, stream>>>(
      z_seq, p_mean, p_std, z_prior, xbuf, xnorm, ynorm);

  const dim3 grid(COL_CHUNKS, ROW_BLOCKS);
  mmd_rbf_tiles<0><<<grid, 256, 0, stream>>>(xbuf,    xbuf,    xnorm, xnorm,
                                             partials + 0 * PARTIALS_PER_MAT);
  mmd_rbf_tiles<1><<<grid, 256, 0, stream>>>(z_prior, z_prior, ynorm, ynorm,
                                             partials + 1 * PARTIALS_PER_MAT);
  mmd_rbf_tiles<2><<<grid, 256, 0, stream>>>(xbuf,    z_prior, xnorm, ynorm,
                                             partials + 2 * PARTIALS_PER_MAT);

  mmd_finalize<<<1, 256, 0, stream>>>(partials, (float*)d_out);
}